// VectorQuantizer_52699248722370
// MI455X (gfx1250) — compile-verified
//
#include <hip/hip_runtime.h>
#include <hip/hip_bf16.h>

typedef float v2f __attribute__((ext_vector_type(2)));
typedef float v8f __attribute__((ext_vector_type(8)));

#define NUM_CODES   1024
#define EMB_DIM     64
#define N_ROWS      32768          // 32*32*32
#define HWSZ        1024           // 32*32
#define CHUNK_CODES 128            // 32KB LDS stage
#define ZQ_ELEMS    2097152        // 32*64*32*32

// ---------------- kernel 0: codebook norms + zero the loss slot ----------------
__global__ void vq_norms_kernel(const float* __restrict__ cb,
                                float* __restrict__ nrm,
                                float* __restrict__ loss_slot) {
    int k = blockIdx.x * blockDim.x + threadIdx.x;
    if (k < NUM_CODES) {
        const float4* p = (const float4*)(cb + k * EMB_DIM);
        float s = 0.0f;
#pragma unroll
        for (int i = 0; i < EMB_DIM / 4; ++i) {
            float4 v = p[i];
            s += v.x * v.x + v.y * v.y + v.z * v.z + v.w * v.w;
        }
        nrm[k] = s;
    }
    if (k == 0) *loss_slot = 0.0f;
}

// ---------------- main kernel: WMMA distance + argmin + writeback ----------------
__global__ __launch_bounds__(256)
void vq_main_kernel(const float* __restrict__ z,
                    const float* __restrict__ cb,
                    const float* __restrict__ nrm,
                    float* __restrict__ zq_out,
                    float* __restrict__ loss_out,
                    float* __restrict__ idx_out) {
    // Operand-order, XOR-swizzled codebook stage (32 KB):
    // element(code, c) -> sop[(c>>1)*256 + ((code*2 + (c&1)) ^ (((c>>1)&1)<<5))]
    __shared__ float sop[CHUNK_CODES * EMB_DIM];
    __shared__ float sck2[CHUNK_CODES];            // staged ||e||^2 for the chunk
    __shared__ int   sidx[8 * 16];                 // per-wave winning indices

    const int tid   = threadIdx.x;
    const int lane  = tid & 31;
    const int wave  = tid >> 5;
    const int lhalf = lane >> 4;       // 0: lanes 0-15, 1: lanes 16-31
    const int lrow  = lane & 15;

    const int gw      = blockIdx.x * 8 + wave;   // global wave id
    const int rowbase = gw * 16;                 // first flat row n
    const int bimg    = rowbase >> 10;           // batch index (HW = 1024)
    const int hwb     = rowbase & (HWSZ - 1);
    const float* zb   = z + (size_t)bimg * (EMB_DIM * HWSZ) + hwb; // + c*HWSZ + r

    // ---- A tile: 16 rows x 64 cols of (-2*z), WMMA 16x4 f32 operand layout ----
    // a[j]: lane<16 holds (row=lrow, k=4j..4j+1); lane>=16 holds k=4j+2..4j+3
    v2f a[16];
#pragma unroll
    for (int j = 0; j < 16; ++j) {
        int c0 = 4 * j + 2 * lhalf;
        float x0 = zb[c0 * HWSZ + lrow];
        float x1 = zb[(c0 + 1) * HWSZ + lrow];
        a[j][0] = -2.0f * x0;
        a[j][1] = -2.0f * x1;
    }

    float bestd[8];
    int   besti[8];
#pragma unroll
    for (int i = 0; i < 8; ++i) { bestd[i] = 3.4e38f; besti[i] = 0; }

    for (int q = 0; q < NUM_CODES / CHUNK_CODES; ++q) {
        __syncthreads();
        // ---- cooperative stage: thread -> code (tid&127), 32 c's (tid>>7) ----
        {
            const int code  = tid & 127;
            const int cbase = (tid >> 7) * 32;
            const float4* src =
                (const float4*)(cb + (size_t)(q * CHUNK_CODES + code) * EMB_DIM + cbase);
#pragma unroll
            for (int g = 0; g < 8; ++g) {
                float4 v = src[g];
                int c = cbase + 4 * g;             // multiple of 4 -> cp0 even
                int cp = c >> 1;
                // pair (c, c+1): swizzle bit = 0 ; pair (c+2, c+3): swizzle bit = 1
                v2f p0 = {v.x, v.y};
                v2f p1 = {v.z, v.w};
                *(v2f*)(sop + cp * 256 + code * 2)              = p0;
                *(v2f*)(sop + (cp + 1) * 256 + ((code * 2) ^ 32)) = p1;
            }
            if (tid < CHUNK_CODES) sck2[tid] = nrm[q * CHUNK_CODES + tid];
        }
        __syncthreads();

        for (int t = 0; t < CHUNK_CODES / 16; ++t) {
            const int codebase = q * CHUNK_CODES + t * 16;
            const int mycode   = codebase + lrow;      // C-layout: N = lane%16
            const int loc      = t * 16 + lrow;        // chunk-local code
            const float ck2    = sck2[loc];
            v8f acc = {ck2, ck2, ck2, ck2, ck2, ck2, ck2, ck2}; // d starts at ||e||^2

            // Conflict-free B operands: one base address, 16-bit DS immediates.
            // bb[j] = {cb[mycode][4j+2h], cb[mycode][4j+2h+1]}
            const v2f* bp = (const v2f*)(sop + lhalf * 256 + ((loc * 2) ^ (lhalf << 5)));
            v2f bb[16];
#pragma unroll
            for (int j = 0; j < 16; ++j) bb[j] = bp[j * 256];

#pragma unroll
            for (int j = 0; j < 16; ++j) {
                acc = __builtin_amdgcn_wmma_f32_16x16x4_f32(
                    false, a[j], false, bb[j], (short)0, acc, false, false);
            }

            // acc[i] = d(row = i + 8*lhalf, code = mycode).
            // Per lane, codes arrive in strictly increasing order, so strict '<'
            // alone preserves lowest-index-on-tie semantics.
#pragma unroll
            for (int i = 0; i < 8; ++i) {
                float d = acc[i];
                bool lt = d < bestd[i];
                bestd[i] = lt ? d : bestd[i];
                besti[i] = lt ? mycode : besti[i];
            }
        }
    }

    // ---- cross-lane argmin: butterfly within each half-wave group of 16 ----
    // Different lanes hold different code residues -> tie-break on lowest index.
#pragma unroll
    for (int i = 0; i < 8; ++i) {
#pragma unroll
        for (int m = 1; m < 16; m <<= 1) {
            float od = __shfl_xor(bestd[i], m, 32);
            int   oi = __shfl_xor(besti[i], m, 32);
            if (od < bestd[i] || (od == bestd[i] && oi < besti[i])) {
                bestd[i] = od; besti[i] = oi;
            }
        }
    }
    // lanes 0-15 hold rows 0..7 winners, lanes 16-31 hold rows 8..15 winners
    if (lane == 0) {
#pragma unroll
        for (int i = 0; i < 8; ++i) sidx[wave * 16 + i] = besti[i];
    }
    if (lane == 16) {
#pragma unroll
        for (int i = 0; i < 8; ++i) sidx[wave * 16 + 8 + i] = besti[i];
    }
    __syncthreads();

    const int myidx = sidx[wave * 16 + lrow];
    if (lane < 16) idx_out[rowbase + lane] = (float)myidx;  // indices as f32

    // ---- z_q writeback (NCHW) + loss accumulation ----
    const float* cbrow = cb + (size_t)myidx * EMB_DIM;
    float* zqb = zq_out + (size_t)bimg * (EMB_DIM * HWSZ) + hwb;
    float lsum = 0.0f;
#pragma unroll
    for (int j = 0; j < 32; ++j) {
        int c = 2 * j + lhalf;
        float vq = cbrow[c];
        float vz = zb[c * HWSZ + lrow];
        float dd = vq - vz;
        lsum += dd * dd;
        zqb[c * HWSZ + lrow] = vq;
    }
#pragma unroll
    for (int m = 1; m < 32; m <<= 1) lsum += __shfl_xor(lsum, m, 32);
    if (lane == 0) {
        // loss = (1 + 0.25) * mean((z_q - z)^2) over 2097152 elements
        atomicAdd(loss_out, lsum * (1.25f / 2097152.0f));
    }
}

extern "C" void kernel_launch(void* const* d_in, const int* in_sizes, int n_in,
                              void* d_out, int out_size, void* d_ws, size_t ws_size,
                              hipStream_t stream) {
    const float* z  = (const float*)d_in[0];
    const float* cb = (const float*)d_in[1];

    float* zq   = (float*)d_out;                 // [32,64,32,32]
    float* loss = (float*)d_out + ZQ_ELEMS;      // scalar
    float* idxo = (float*)d_out + ZQ_ELEMS + 1;  // [32768]
    float* nrm  = (float*)d_ws;                  // 1024 f32 codebook norms

    vq_norms_kernel<<<4, 256, 0, stream>>>(cb, nrm, loss);
    vq_main_kernel<<<256, 256, 0, stream>>>(z, cb, nrm, zq, loss, idxo);
}